// GCN_23295902613546
// MI455X (gfx1250) — compile-verified
//
#include <hip/hip_runtime.h>

// ---------------------------------------------------------------------------
// GCN forward for MI455X (gfx1250, wave32).
//   h  = relu(spmm(A, x @ W_in))
//   h  = relu(spmm(A, h @ W_h1)) + h
//   h  = relu(spmm(A, h @ W_h2)) + h
//   out= spmm(A, h @ W_out)
// Dense GEMMs: v_wmma_f32_16x16x4_f32; one wave computes a 32x64 output tile
// (2 M-tiles x 4 N-tiles, 8 accumulators) so each K-step issues 10 loads for
// 8 WMMAs, and A/B fragments are multiply reused.
// SpMM: coalesced gather + global_atomic_add_f32 scatter; the 102 MB feature
// matrix is L2-resident (192 MB L2), so gathers/atomics mostly hit L2.
// ---------------------------------------------------------------------------

typedef __attribute__((ext_vector_type(2))) float v2f;
typedef __attribute__((ext_vector_type(8))) float v8f;

#define N_NODES 100000
#define N_EDGES 3200000
#define F_HID   256
#define F_OUT   40
#define F_OUTP  48     // F_OUT padded to 3x16 tiles

// ---------------- elementwise helpers (float4 for bandwidth) ----------------

__global__ void zero_f4(float4* __restrict__ p, int n4) {
    int i = blockIdx.x * blockDim.x + threadIdx.x;
    if (i < n4) p[i] = make_float4(0.f, 0.f, 0.f, 0.f);
}

__global__ void relu_k(float4* __restrict__ H, const float4* __restrict__ S, int n4) {
    int i = blockIdx.x * blockDim.x + threadIdx.x;
    if (i < n4) {
        float4 s = S[i];
        H[i] = make_float4(fmaxf(s.x, 0.f), fmaxf(s.y, 0.f),
                           fmaxf(s.z, 0.f), fmaxf(s.w, 0.f));
    }
}

__global__ void relu_res_k(float4* __restrict__ H, const float4* __restrict__ S, int n4) {
    int i = blockIdx.x * blockDim.x + threadIdx.x;
    if (i < n4) {
        float4 s = S[i];
        float4 h = H[i];
        H[i] = make_float4(fmaxf(s.x, 0.f) + h.x, fmaxf(s.y, 0.f) + h.y,
                           fmaxf(s.z, 0.f) + h.z, fmaxf(s.w, 0.f) + h.w);
    }
}

// Zero-pad W_out [256,40] -> [256,48] so the head GEMM needs no load guards.
__global__ void pad_wout(const float* __restrict__ W, float* __restrict__ Wp) {
    int i = blockIdx.x * blockDim.x + threadIdx.x;   // 256*48 threads
    int r = i / F_OUTP, c = i % F_OUTP;
    Wp[i] = (c < F_OUT) ? W[r * F_OUT + c] : 0.f;
}

// ---------------- dense GEMM: O[M,Nout] = A[M,K] @ W[K,Nw] ------------------
// One wave32 computes a 32 x (16*NT) output tile via V_WMMA_F32_16X16X4_F32.
//
// ISA VGPR layouts (CDNA5 ISA):
//   A 16x4 (32-bit): lane = (k>>1)*16 + m, vgpr = k&1  -> per-lane v2f along K
//   B  4x16 (32-bit): lane = (k>>1)*16 + n, vgpr = k&1
//   C/D 16x16 f32:    vgpr v: lanes 0-15 -> M=v, N=lane; lanes 16-31 -> M=v+8
//
// M is a multiple of 32 (100000 = 3125*32). Nw (the stride of W) covers whole
// tiles (W is padded), so the K loop has no guards; only stores check Nout.
// Per K-step: 2 A loads (b64) + 2*NT B loads (b32) -> 2*NT WMMAs.
template <int NT>
__launch_bounds__(128)
__global__ void gemm_wmma_f32(const float* __restrict__ A,
                              const float* __restrict__ W,   // [K, Nw], padded
                              float* __restrict__ O,         // [M, Nout]
                              int Nw, int Nout, int K) {
    const int lane  = threadIdx.x & 31;
    const int wave  = threadIdx.x >> 5;
    const int m0    = blockIdx.x << 5;                           // 32 rows/block
    const int n0    = (blockIdx.y * (blockDim.x >> 5) + wave) * (16 * NT);
    const int nlane = lane & 15;
    const int koff  = (lane >> 4) << 1;        // lanes 0-15: K{0,1}; 16-31: K{2,3}

    const float* __restrict__ ap0 = A + (m0 + (lane & 15)) * K + koff;
    const float* __restrict__ ap1 = ap0 + 16 * K;
    const float* __restrict__ bp0 = W + koff * Nw + n0 + nlane;   // row k+koff
    const float* __restrict__ bp1 = bp0 + Nw;                     // row k+koff+1
    const int bstep = 4 * Nw;

    v8f acc0[NT], acc1[NT];
#pragma unroll
    for (int t = 0; t < NT; ++t) {
        acc0[t] = (v8f){0.f,0.f,0.f,0.f,0.f,0.f,0.f,0.f};
        acc1[t] = (v8f){0.f,0.f,0.f,0.f,0.f,0.f,0.f,0.f};
    }

    for (int k = 0; k < K; k += 4) {
        // Hoist all loads for this K-step so they clause together and one
        // wait covers the whole WMMA burst.
        v2f a0 = *(const v2f*)ap0;             // 8-byte aligned (koff even)
        v2f a1 = *(const v2f*)ap1;
        v2f b[NT];
#pragma unroll
        for (int t = 0; t < NT; ++t) {
            b[t].x = bp0[t * 16];              // immediate-offset loads
            b[t].y = bp1[t * 16];
        }
#pragma unroll
        for (int t = 0; t < NT; ++t)
            acc0[t] = __builtin_amdgcn_wmma_f32_16x16x4_f32(
                false, a0, false, b[t], (short)0, acc0[t], false, false);
#pragma unroll
        for (int t = 0; t < NT; ++t)
            acc1[t] = __builtin_amdgcn_wmma_f32_16x16x4_f32(
                false, a1, false, b[t], (short)0, acc1[t], false, false);
        ap0 += 4;
        ap1 += 4;
        bp0 += bstep;
        bp1 += bstep;
    }

    const int cm0 = m0 + ((lane >> 4) << 3);   // +8 for lanes 16-31
    const int cm1 = cm0 + 16;
#pragma unroll
    for (int t = 0; t < NT; ++t) {
        const int cn = n0 + t * 16 + nlane;
        if (cn < Nout) {
#pragma unroll
            for (int v = 0; v < 8; ++v) {
                O[(cm0 + v) * Nout + cn] = acc0[t][v];
                O[(cm1 + v) * Nout + cn] = acc1[t][v];
            }
        }
    }
}

// ---------------- SpMM scatter: O[rows[e],:] += vals[e] * X[cols[e],:] -------
__launch_bounds__(256)
__global__ void spmm_scatter_f256(const int*   __restrict__ rows,
                                  const int*   __restrict__ cols,
                                  const float* __restrict__ vals,
                                  const float* __restrict__ X,
                                  float* __restrict__ O) {
    const int e = blockIdx.x;
    const int f = threadIdx.x;
    const int r = rows[e];
    const int c = cols[e];
    const float v = vals[e];
    atomicAdd(O + r * F_HID + f, v * X[c * F_HID + f]);
}

__launch_bounds__(64)
__global__ void spmm_scatter_f40(const int*   __restrict__ rows,
                                 const int*   __restrict__ cols,
                                 const float* __restrict__ vals,
                                 const float* __restrict__ X,
                                 float* __restrict__ O) {
    const int e = blockIdx.x;
    const int f = threadIdx.x;
    if (f >= F_OUT) return;
    const int r = rows[e];
    const int c = cols[e];
    const float v = vals[e];
    atomicAdd(O + r * F_OUT + f, v * X[c * F_OUT + f]);
}

// ---------------------------------------------------------------------------

extern "C" void kernel_launch(void* const* d_in, const int* in_sizes, int n_in,
                              void* d_out, int out_size, void* d_ws, size_t ws_size,
                              hipStream_t stream) {
    const float* x    = (const float*)d_in[0];   // [100000, 512]
    const int*   er   = (const int*)  d_in[1];   // [3.2M]
    const int*   ec   = (const int*)  d_in[2];   // [3.2M]
    const float* ev   = (const float*)d_in[3];   // [3.2M]
    const float* Win  = (const float*)d_in[4];   // [512, 256]
    const float* Wh1  = (const float*)d_in[5];   // [256, 256]
    const float* Wh2  = (const float*)d_in[6];   // [256, 256]
    const float* Wout = (const float*)d_in[7];   // [256, 40]
    float* out = (float*)d_out;                  // [100000, 40]

    const long long HID = (long long)N_NODES * F_HID;   // 25.6M floats
    float* A    = (float*)d_ws;     // h           [100000, 256]
    float* B    = A + HID;          // h @ W       [100000, 256]
    float* C    = B + HID;          // spmm accum  [100000, 256]; reused as
                                    // [100000, 40] head-GEMM output
    float* Wpad = C + HID;          // W_out padded [256, 48]

    const int HID4 = (int)(HID / 4);                     // 6.4M float4
    const int OUT4 = (N_NODES * F_OUT) / 4;              // 1.0M float4
    const dim3 gEl((HID4 + 255) / 256);
    const dim3 gOut((OUT4 + 255) / 256);
    // 4 waves/block, each wave = 32x64 tile -> one block covers 32x256.
    const dim3 gGemm(N_NODES / 32, 1);                   // 3125 blocks

    // ---- layer 1: h = relu(spmm(x @ W_in)) ----
    gemm_wmma_f32<4><<<gGemm, 128, 0, stream>>>(x, Win, B, F_HID, F_HID, 512);
    zero_f4<<<gEl, 256, 0, stream>>>((float4*)C, HID4);
    spmm_scatter_f256<<<N_EDGES, 256, 0, stream>>>(er, ec, ev, B, C);
    relu_k<<<gEl, 256, 0, stream>>>((float4*)A, (const float4*)C, HID4);

    // ---- layer 2: h = relu(spmm(h @ W_h1)) + h ----
    gemm_wmma_f32<4><<<gGemm, 128, 0, stream>>>(A, Wh1, B, F_HID, F_HID, F_HID);
    zero_f4<<<gEl, 256, 0, stream>>>((float4*)C, HID4);
    spmm_scatter_f256<<<N_EDGES, 256, 0, stream>>>(er, ec, ev, B, C);
    relu_res_k<<<gEl, 256, 0, stream>>>((float4*)A, (const float4*)C, HID4);

    // ---- layer 3: h = relu(spmm(h @ W_h2)) + h ----
    gemm_wmma_f32<4><<<gGemm, 128, 0, stream>>>(A, Wh2, B, F_HID, F_HID, F_HID);
    zero_f4<<<gEl, 256, 0, stream>>>((float4*)C, HID4);
    spmm_scatter_f256<<<N_EDGES, 256, 0, stream>>>(er, ec, ev, B, C);
    relu_res_k<<<gEl, 256, 0, stream>>>((float4*)A, (const float4*)C, HID4);

    // ---- head: out = spmm(h @ W_out) ----
    pad_wout<<<(256 * F_OUTP) / 256, 256, 0, stream>>>(Wout, Wpad);
    gemm_wmma_f32<3><<<gGemm, 32, 0, stream>>>(A, Wpad, C, F_OUTP, F_OUT, F_HID);
    zero_f4<<<gOut, 256, 0, stream>>>((float4*)out, OUT4);
    spmm_scatter_f40<<<N_EDGES, 64, 0, stream>>>(er, ec, ev, C, out);
}